// SemCHGraphConv_30502857736584
// MI455X (gfx1250) — compile-verified
//
#include <hip/hip_runtime.h>

typedef __attribute__((ext_vector_type(2))) float v2f;
typedef __attribute__((ext_vector_type(4))) float v4f;
typedef __attribute__((ext_vector_type(8))) float v8f;

typedef int v4i_ __attribute__((vector_size(16)));
typedef __attribute__((address_space(1))) v4i_* gptr_v4i;
typedef __attribute__((address_space(3))) v4i_* lptr_v4i;

#define NJ    17      // joints
#define NC    128     // out channels
#define NFIN  128     // in features
#define NNZ   49      // sparse adjacency entries
#define FR    6       // frames per block
#define ROWS  (FR*NJ) // 102 valid rows per block
#define MT    7       // M-tiles of 16 (padded to 112)
#define PR    (MT*16) // 112 padded rows
#define XP    132     // LDS row pitch (floats): bank = (4*row+k)%64 -> conflict-free

#if defined(__gfx1250__) && __has_builtin(__builtin_amdgcn_global_load_async_to_lds_b128)
#define HAVE_ASYNC_LDS 1
#else
#define HAVE_ASYNC_LDS 0
#endif

// ---------------------------------------------------------------------------
// Pass 1: per-channel softmax of sparse adjacency -> normalized weights p[c][k]
// plus packed per-joint neighbor lists (shared across channels).
// ---------------------------------------------------------------------------
__global__ void semgcn_setup(const float* __restrict__ e,
                             const int* __restrict__ idx_i,
                             const int* __restrict__ idx_j,
                             float* __restrict__ p,
                             int* __restrict__ nbrcnt,
                             int* __restrict__ nbrpk)
{
    __shared__ float ex[64];
    __shared__ float rowsum[NJ];
    const int c = blockIdx.x;
    const int t = threadIdx.x;

    if (t < NNZ) ex[t] = expf(e[c * NNZ + t]);
    __syncthreads();
    if (t < NJ) {
        float s = 0.f;
        for (int k = 0; k < NNZ; ++k)
            if (idx_i[k] == t) s += ex[k];
        rowsum[t] = s;
    }
    __syncthreads();
    if (t < NNZ) p[c * NNZ + t] = ex[t] / rowsum[idx_i[t]];

    if (c == 0 && t < NJ) {
        int cnt = 0;
        for (int k = 0; k < NNZ; ++k) {
            if (idx_i[k] == t) {
                int nj = idx_j[k];
                nbrpk[t * 8 + cnt] = nj | (k << 8) | ((nj == t) ? (1 << 16) : 0);
                ++cnt;
            }
        }
        nbrcnt[t] = cnt;
    }
}

// ---------------------------------------------------------------------------
// Pass 2: fused dual-GEMM (WMMA f32 16x16x4) + graph aggregation.
// Block = 6 frames (102 rows), 8 waves; wave w owns channel tile [16w,16w+16).
// x tile staged via CDNA5 async global->LDS DMA (no VGPR round-trip).
// ---------------------------------------------------------------------------
__global__ __launch_bounds__(256)
void semgcn_main(const float* __restrict__ x,
                 const float* __restrict__ W,
                 const float* __restrict__ p,
                 const int* __restrict__ nbrcnt,
                 const int* __restrict__ nbrpk,
                 float* __restrict__ out,
                 int bt_total)
{
    __shared__ float xs [PR * XP];   // x tile, pitch-padded
    __shared__ float h1s[PR * XP];   // h1 tile for aggregation
    __shared__ float ps [NC * NNZ];  // softmax weights
    __shared__ int   s_cnt[NJ];
    __shared__ int   s_pk [NJ * 8];

    const int tid  = threadIdx.x;
    const int wave = tid >> 5;
    const int lane = tid & 31;
    const int half = lane >> 4;   // K-half for A/B fragments, M-half for C/D
    const int l15  = lane & 15;

    const long long rowBase   = (long long)blockIdx.x * ROWS;
    const long long totalRows = (long long)bt_total * NJ;

    // ---- stage x tile (padded rows loaded too; they are other blocks' valid rows) ----
#if HAVE_ASYNC_LDS
    for (int idx = tid; idx < PR * (NFIN / 4); idx += 256) {
        const int r  = idx >> 5;              // NFIN/4 == 32 float4's per row
        const int k4 = (idx & 31) * 4;
        const long long grow = rowBase + r;
        if (grow < totalRows) {
            __builtin_amdgcn_global_load_async_to_lds_b128(
                (gptr_v4i)(x + grow * NFIN + k4),
                (lptr_v4i)(&xs[r * XP + k4]),
                0, 0);
        }
    }
#else
    for (int idx = tid; idx < PR * (NFIN / 4); idx += 256) {
        const int r  = idx / (NFIN / 4);
        const int k4 = (idx % (NFIN / 4)) * 4;
        const long long grow = rowBase + r;
        v4f v = {0.f, 0.f, 0.f, 0.f};
        if (grow < totalRows) v = *(const v4f*)(x + grow * NFIN + k4);
        *(v4f*)(&xs[r * XP + k4]) = v;
    }
#endif
    for (int idx = tid; idx < NC * NNZ; idx += 256) ps[idx] = p[idx];
    if (tid < NJ)     s_cnt[tid] = nbrcnt[tid];
    if (tid < NJ * 8) s_pk[tid]  = nbrpk[tid];
#if HAVE_ASYNC_LDS
#if __has_builtin(__builtin_amdgcn_s_wait_asynccnt)
    __builtin_amdgcn_s_wait_asynccnt(0);
#else
    asm volatile("s_wait_asynccnt 0" ::: "memory");
#endif
#endif
    __syncthreads();

    const int nb = wave * 16;      // channel-tile base
    const int ch = nb + l15;       // this lane's channel (C/D: N = lane&15)

    // B-fragment register block: 128x16 slice of W, 16x16x4 layout:
    //   lanes 0-15 : K = 4s, 4s+1 ; lanes 16-31 : K = 4s+2, 4s+3 ; N = lane&15
    v2f bf[32];

    // ================= phase 1: h1 = x * W[1] -> LDS =================
    {
        const float* Wm = W + 1 * NFIN * NC;
#pragma unroll
        for (int s = 0; s < 32; ++s) {
            const int k = 4 * s + 2 * half;
            bf[s].x = Wm[(k + 0) * NC + ch];
            bf[s].y = Wm[(k + 1) * NC + ch];
        }
    }
    for (int mt = 0; mt < MT; ++mt) {
        v8f acc = {};
#pragma unroll
        for (int s = 0; s < 32; ++s) {
            const int k = 4 * s + 2 * half;
            const v2f a = *(const v2f*)(&xs[(mt * 16 + l15) * XP + k]); // A: M=lane&15
            acc = __builtin_amdgcn_wmma_f32_16x16x4_f32(
                false, a, false, bf[s], (short)0, acc, false, false);
        }
#pragma unroll
        for (int i = 0; i < 8; ++i) {
            const int r = mt * 16 + i + 8 * half;   // C/D: VGPR i -> M = i + 8*half
            if (r < ROWS) h1s[r * XP + ch] = acc[i];
        }
    }
    __syncthreads();

    // ================= phase 2: h0 = x * W[0], aggregate, store =================
    {
        const float* Wm = W;  // W[0]
#pragma unroll
        for (int s = 0; s < 32; ++s) {
            const int k = 4 * s + 2 * half;
            bf[s].x = Wm[(k + 0) * NC + ch];
            bf[s].y = Wm[(k + 1) * NC + ch];
        }
    }
    for (int mt = 0; mt < MT; ++mt) {
        v8f acc = {};
#pragma unroll
        for (int s = 0; s < 32; ++s) {
            const int k = 4 * s + 2 * half;
            const v2f a = *(const v2f*)(&xs[(mt * 16 + l15) * XP + k]);
            acc = __builtin_amdgcn_wmma_f32_16x16x4_f32(
                false, a, false, bf[s], (short)0, acc, false, false);
        }
#pragma unroll
        for (int i = 0; i < 8; ++i) {
            const int r = mt * 16 + i + 8 * half;   // local row (uniform per half-group)
            if (r >= ROWS) continue;
            const long long grow = rowBase + r;
            if (grow >= totalRows) continue;
            const int f = r / NJ;
            const int j = r % NJ;
            float res = 0.f;
            const int cnt = s_cnt[j];
            for (int t = 0; t < cnt; ++t) {
                const int pk     = s_pk[j * 8 + t];
                const int njt    = pk & 0xFF;
                const int kidx   = (pk >> 8) & 0xFF;
                const int isdiag = pk >> 16;
                const float pw = ps[ch * NNZ + kidx];
                const float hv = isdiag ? acc[i] : h1s[(f * NJ + njt) * XP + ch];
                res += pw * hv;
            }
            out[grow * NC + ch] = res;
        }
    }
}

// ---------------------------------------------------------------------------
extern "C" void kernel_launch(void* const* d_in, const int* in_sizes, int n_in,
                              void* d_out, int out_size, void* d_ws, size_t ws_size,
                              hipStream_t stream)
{
    const float* x     = (const float*)d_in[0];
    const float* W     = (const float*)d_in[1];
    const float* e     = (const float*)d_in[2];
    const int*   idx_i = (const int*)d_in[3];
    const int*   idx_j = (const int*)d_in[4];
    float*       out   = (float*)d_out;

    const int bt_total = in_sizes[0] / (NJ * NFIN);   // 64*243 = 15552

    // workspace layout: p[NC*NNZ] | nbrcnt[NJ] | nbrpk[NJ*8]
    float* p      = (float*)d_ws;
    int*   nbrcnt = (int*)((char*)d_ws + (size_t)NC * NNZ * sizeof(float));
    int*   nbrpk  = nbrcnt + NJ;

    semgcn_setup<<<NC, 64, 0, stream>>>(e, idx_i, idx_j, p, nbrcnt, nbrpk);

    const int blocks = (bt_total + FR - 1) / FR;      // 2592
    semgcn_main<<<blocks, 256, 0, stream>>>(x, W, p, nbrcnt, nbrpk, out, bt_total);
}